// GFCN_55559696941160
// MI455X (gfx1250) — compile-verified
//
#include <hip/hip_runtime.h>
#include <hip/hip_bf16.h>
#include <math.h>

typedef float v2f __attribute__((ext_vector_type(2)));
typedef float v8f __attribute__((ext_vector_type(8)));

// ---------------- spline basis: degree-1, K=5 per dim, 4 (idx,w) pairs ----------------
__device__ __forceinline__ void spline4(float a0, float a1, int idx[4], float w[4]) {
    float v0 = a0 * 4.0f, v1 = a1 * 4.0f;
    float l0 = fminf(fmaxf(floorf(v0), 0.0f), 3.0f);
    float l1 = fminf(fmaxf(floorf(v1), 0.0f), 3.0f);
    float t0 = v0 - l0, t1 = v1 - l1;
    int i0 = (int)l0, i1 = (int)l1;
    idx[0] = i0 * 5 + i1;           w[0] = (1.f - t0) * (1.f - t1);
    idx[1] = i0 * 5 + i1 + 1;       w[1] = (1.f - t0) * t1;
    idx[2] = (i0 + 1) * 5 + i1;     w[2] = t0 * (1.f - t1);
    idx[3] = (i0 + 1) * 5 + i1 + 1; w[3] = t0 * t1;
}

// ---------------- f32 WMMA GEMM: C[M,N] = A[M,K] @ B[K,N] ----------------
// A row-major with leading dim lda; B supplied TRANSPOSED as Bt[N][K] so each
// lane's fragment is a contiguous K-pair -> unconditional global_load_b64.
// Requires: M%64==0, N%16==0, Kd%4==0 (caller pads workspace to guarantee it).
// V_WMMA_F32_16X16X4_F32 lane layout (ISA 7.12.2):
//   A/B: element [row|col = lane&15][k = 2*(lane>>4) + v] in VGPR v
//   D:   element r -> row (8*(lane>>4) + r), col (lane&15)
__global__ __launch_bounds__(128) void k_gemm_wmma_f32(
    const float* __restrict__ A, int lda,
    const float* __restrict__ Bt,
    float* __restrict__ C, int N, int Kd)
{
    const int lane   = threadIdx.x;        // 0..31
    const int half   = lane >> 4;
    const int lane16 = lane & 15;
    const int tileM  = blockIdx.x * 4 + threadIdx.y;
    const int m      = tileM * 16 + lane16;          // A row for this lane
    const int n      = blockIdx.y * 16 + lane16;     // B col / C col for this lane
    const float* ap  = A  + (size_t)m * lda + 2 * half;
    const float* bp  = Bt + (size_t)n * Kd + 2 * half;
    v8f acc = {0.f, 0.f, 0.f, 0.f, 0.f, 0.f, 0.f, 0.f};
    for (int kb = 0; kb < Kd; kb += 4) {
        v2f av = *(const v2f*)(ap + kb);   // 8B aligned: lda,Kd even, kb%4==0
        v2f bv = *(const v2f*)(bp + kb);
        acc = __builtin_amdgcn_wmma_f32_16x16x4_f32(
            false, av, false, bv, (short)0, acc, false, false);
    }
    const int rbase = tileM * 16 + 8 * half;
    float* cp = C + (size_t)rbase * N + n;
    #pragma unroll
    for (int r = 0; r < 8; ++r)
        cp[(size_t)r * N] = acc[r];
}

// ---------------- B packs (tiny; W is [25, Cin, Cout] row-major) ----------------
// G-side: B row index k = kk*Cin + i, so Bt[o*Kp + kk*Cin + i] = W[kk,i,o]
__global__ void k_pack_g(const float* __restrict__ W, int Cin, int Cout, int Kp,
                         float* __restrict__ Bt)
{
    int t = blockIdx.x * blockDim.x + threadIdx.x;
    int tot = 25 * Cin * Cout;
    if (t >= tot) return;
    int kk = t / (Cin * Cout);
    int r  = t % (Cin * Cout);
    int i  = r / Cout, o = r % Cout;
    Bt[(size_t)o * Kp + kk * Cin + i] = W[t];
}

// Z-side: output col j = kk*Cout + o over K=Cin, so Bt[j*Cin + i] = W[kk,i,o]
__global__ void k_pack_z(const float* __restrict__ W, int Cin, int Cout,
                         float* __restrict__ Bt)
{
    int t = blockIdx.x * blockDim.x + threadIdx.x;
    int tot = 25 * Cin * Cout;
    if (t >= tot) return;
    int kk = t / (Cin * Cout);
    int r  = t % (Cin * Cout);
    int i  = r / Cout, o = r % Cout;
    Bt[(size_t)(kk * Cout + o) * Cin + i] = W[t];
}

// ---------------- G-side edge scatter: G[dst*Gs + idx_p*Cin + i] += w_p * x[src, i] ----------------
__global__ void k_edge_scatter_G(const float* __restrict__ x, int Cin, int Gs,
                                 const int* __restrict__ src, const int* __restrict__ dst,
                                 const float* __restrict__ ea, int E,
                                 float* __restrict__ G, float* __restrict__ cnt)
{
    int e = blockIdx.x * blockDim.x + threadIdx.x;
    if (e >= E) return;
    int s = src[e], d = dst[e];
    int idx[4]; float w[4];
    spline4(ea[2 * e], ea[2 * e + 1], idx, w);
    atomicAdd(&cnt[d], 1.0f);
    size_t gbase = (size_t)d * Gs;
    size_t xbase = (size_t)s * Cin;
    for (int p = 0; p < 4; ++p) {
        size_t b = gbase + (size_t)idx[p] * Cin;
        float wp = w[p];
        for (int i = 0; i < Cin; ++i)
            atomicAdd(&G[b + i], wp * x[xbase + i]);
    }
}

// ---------------- Z-side edge gather: agg[dst] += sum_p w_p * Z[src*ldZ + idx_p*Cout + :] ----------------
__global__ void k_edge_gather_Z(const float* __restrict__ Z, int Cout, int ldZ,
                                const int* __restrict__ src, const int* __restrict__ dst,
                                const float* __restrict__ ea, int E,
                                float* __restrict__ agg, float* __restrict__ cnt)
{
    int e = blockIdx.x * blockDim.x + threadIdx.x;
    if (e >= E) return;
    int s = src[e], d = dst[e];
    int idx[4]; float w[4];
    spline4(ea[2 * e], ea[2 * e + 1], idx, w);
    atomicAdd(&cnt[d], 1.0f);
    size_t zbase = (size_t)s * ldZ;
    float msg[32];
    for (int o = 0; o < Cout; ++o) msg[o] = 0.f;
    for (int p = 0; p < 4; ++p) {
        const float* zp = Z + zbase + (size_t)idx[p] * Cout;
        float wp = w[p];
        for (int o = 0; o < Cout; ++o) msg[o] += wp * zp[o];
    }
    size_t ab = (size_t)d * Cout;
    for (int o = 0; o < Cout; ++o) atomicAdd(&agg[ab + o], msg[o]);
}

// ---------------- out = ELU(agg/max(cnt,1) + X@root + bias) ----------------
__global__ void k_epilogue_elu(const float* __restrict__ agg, const float* __restrict__ cnt,
                               const float* __restrict__ X, int Cin,
                               const float* __restrict__ root, const float* __restrict__ bias,
                               float* __restrict__ out, int N, int Cout)
{
    long t = (long)blockIdx.x * blockDim.x + threadIdx.x;
    if (t >= (long)N * Cout) return;
    int n = (int)(t / Cout), c = (int)(t % Cout);
    float v = agg[t] / fmaxf(cnt[n], 1.0f);
    const float* xr = X + (size_t)n * Cin;
    float acc = 0.f;
    for (int i = 0; i < Cin; ++i) acc += xr[i] * root[(size_t)i * Cout + c];
    v += acc + bias[c];
    out[t] = (v > 0.f) ? v : expm1f(v);
}

// ---------------- segment helpers ----------------
__global__ void k_count(const int* __restrict__ cl, int N, float* __restrict__ cnt)
{
    int t = blockIdx.x * blockDim.x + threadIdx.x;
    if (t >= N) return;
    atomicAdd(&cnt[cl[t]], 1.0f);
}

__global__ void k_scatter_feat(const float* __restrict__ v, const int* __restrict__ cl,
                               int N, int C, float* __restrict__ sum)
{
    long t = (long)blockIdx.x * blockDim.x + threadIdx.x;
    if (t >= (long)N * C) return;
    int n = (int)(t / C), c = (int)(t % C);
    atomicAdd(&sum[(size_t)cl[n] * C + c], v[t]);
}

__global__ void k_div_cnt(float* __restrict__ buf, const float* __restrict__ cnt, int N, int C)
{
    long t = (long)blockIdx.x * blockDim.x + threadIdx.x;
    if (t >= (long)N * C) return;
    int n = (int)(t / C);
    buf[t] = buf[t] / fmaxf(cnt[n], 1.0f);
}

__global__ void k_pweights(const float* __restrict__ x, const int* __restrict__ cl,
                           int N, int C, const float* __restrict__ ccnt,
                           const float* __restrict__ gsum, float* __restrict__ out)
{
    long t = (long)blockIdx.x * blockDim.x + threadIdx.x;
    if (t >= (long)N * C) return;
    int n = (int)(t / C), c = (int)(t % C);
    int k = cl[n];
    float r = ccnt[k] * x[t] / (gsum[(size_t)k * C + c] + 0.001f);
    out[t] = (r == r) ? r : 0.0f;   // NaN -> 0
}

__global__ void k_map_edges(const int* __restrict__ ei, const int* __restrict__ cl,
                            long n, int* __restrict__ out)
{
    long t = (long)blockIdx.x * blockDim.x + threadIdx.x;
    if (t >= n) return;
    out[t] = cl[ei[t]];
}

// ---------------- cartesian edge attributes with global max ----------------
__global__ void k_cart_d(const float* __restrict__ pos, const int* __restrict__ src,
                         const int* __restrict__ dst, int E,
                         float* __restrict__ ea, float* __restrict__ red)
{
    int e = blockIdx.x * blockDim.x + threadIdx.x;
    if (e >= E) return;
    int s = src[e], d = dst[e];
    float dx = pos[2 * (size_t)d]     - pos[2 * (size_t)s];
    float dy = pos[2 * (size_t)d + 1] - pos[2 * (size_t)s + 1];
    ea[2 * (size_t)e] = dx;
    ea[2 * (size_t)e + 1] = dy;
    float m = fmaxf(fabsf(dx), fabsf(dy));   // non-negative: int compare == float compare
    atomicMax((int*)red, __float_as_int(m));
}

__global__ void k_cart_norm(float* __restrict__ ea, const float* __restrict__ red, long n2)
{
    long t = (long)blockIdx.x * blockDim.x + threadIdx.x;
    if (t >= n2) return;
    float mx = red[0];
    float s = 1.0f / (2.0f * mx + 1e-12f);
    ea[t] = ea[t] * s + 0.5f;
}

// ---------------- barycentric upsample: out[n,c] = h[cl[n],c] * w[n, (wC==1?0:c)] ----------------
__global__ void k_upsample(const float* __restrict__ h, const int* __restrict__ cl,
                           const float* __restrict__ w, int wC,
                           int N, int C, float* __restrict__ out)
{
    long t = (long)blockIdx.x * blockDim.x + threadIdx.x;
    if (t >= (long)N * C) return;
    int n = (int)(t / C), c = (int)(t % C);
    out[t] = h[(size_t)cl[n] * C + c] * w[(size_t)n * wC + (wC == 1 ? 0 : c)];
}

extern "C" void kernel_launch(void* const* d_in, const int* in_sizes, int n_in,
                              void* d_out, int out_size, void* d_ws, size_t ws_size,
                              hipStream_t stream)
{
    (void)n_in; (void)out_size; (void)ws_size;
    const float* x     = (const float*)d_in[0];
    const float* pos   = (const float*)d_in[1];
    const float* eattr = (const float*)d_in[2];
    const int*   ei    = (const int*)d_in[3];
    const int*   cl1   = (const int*)d_in[4];
    const int*   cl2   = (const int*)d_in[5];
    const float* W1 = (const float*)d_in[6],  *r1 = (const float*)d_in[7],  *b1 = (const float*)d_in[8];
    const float* W2 = (const float*)d_in[9],  *r2 = (const float*)d_in[10], *b2 = (const float*)d_in[11];
    const float* W3 = (const float*)d_in[12], *r3 = (const float*)d_in[13], *b3 = (const float*)d_in[14];
    const float* W4 = (const float*)d_in[15], *r4 = (const float*)d_in[16], *b4 = (const float*)d_in[17];
    const float* W5 = (const float*)d_in[18], *r5 = (const float*)d_in[19], *b5 = (const float*)d_in[20];

    const int N1 = in_sizes[0];
    const int E  = in_sizes[3] / 2;
    const int N2 = in_sizes[5];
    const int N3 = N2 / 2;

    // ---- workspace carve (bump allocator, 256B aligned) ----
    char* ws = (char*)d_ws;
    size_t off = 0;
    auto carve = [&](size_t nfloats) -> float* {
        size_t o = (off + 255) & ~(size_t)255;
        off = o + nfloats * sizeof(float);
        return (float*)(ws + o);
    };
    float* bufA = carve((size_t)N1 * 32);   // h1 / h2 / hb1 (sequentially dead -> shared)
    float* w1b  = carve((size_t)N1);
    float* x2   = carve((size_t)N2 * 32);
    float* pos2 = carve((size_t)N2 * 2);
    int*   ei2  = (int*)carve((size_t)2 * E);
    float* ea2  = carve((size_t)2 * E);
    float* w2b  = carve((size_t)N2 * 32);
    float* x3   = carve((size_t)N3 * 64);
    float* pos3 = carve((size_t)N3 * 2);
    int*   ei3  = (int*)carve((size_t)2 * E);
    float* ea3  = carve((size_t)2 * E);
    float* h3   = carve((size_t)N3 * 32);
    float* hb2  = carve((size_t)N2 * 32);
    float* h4   = carve((size_t)N2 * 32);
    float* ea1c = carve((size_t)2 * E);
    float* cnt  = carve((size_t)N1);
    float* cc1  = carve((size_t)N2);
    float* gs1  = carve((size_t)N2);
    float* cc2  = carve((size_t)N3);
    float* gs2  = carve((size_t)N3 * 32);
    float* red  = carve(8);
    float* wpk  = carve((size_t)800 * 64); // packed Bt, max 51200 floats
    float* agg  = carve((size_t)N1 * 32);
    float* bigG = carve((size_t)N2 * 800); // reused: G1(pad28), G2, Z3, Z4, Z5(pad32)

    const int TB = 256;
    auto nb = [](long n) { return (unsigned)((n + 255) / 256); };

    // ======== conv1 (G-side, Cin=1 -> Cout=32; K padded 25 -> 28) ========
    hipMemsetAsync(bigG, 0, (size_t)N1 * 28 * 4, stream);
    hipMemsetAsync(cnt,  0, (size_t)N1 * 4, stream);
    k_edge_scatter_G<<<nb(E), TB, 0, stream>>>(x, 1, 28, ei, ei + E, eattr, E, bigG, cnt);
    hipMemsetAsync(wpk, 0, (size_t)32 * 28 * 4, stream);
    k_pack_g<<<nb(25 * 1 * 32), TB, 0, stream>>>(W1, 1, 32, 28, wpk);
    k_gemm_wmma_f32<<<dim3(N1 / 64, 2), dim3(32, 4), 0, stream>>>(bigG, 28, wpk, agg, 32, 28);
    k_epilogue_elu<<<nb((long)N1 * 32), TB, 0, stream>>>(agg, cnt, x, 1, r1, b1, bufA, N1, 32);

    // ======== pool level1 -> level2 ========
    hipMemsetAsync(cc1, 0, (size_t)N2 * 4, stream);
    k_count<<<nb(N1), TB, 0, stream>>>(cl1, N1, cc1);
    hipMemsetAsync(gs1, 0, (size_t)N2 * 4, stream);
    k_scatter_feat<<<nb(N1), TB, 0, stream>>>(x, cl1, N1, 1, gs1);
    k_pweights<<<nb(N1), TB, 0, stream>>>(x, cl1, N1, 1, cc1, gs1, w1b);
    hipMemsetAsync(x2, 0, (size_t)N2 * 32 * 4, stream);
    k_scatter_feat<<<nb((long)N1 * 32), TB, 0, stream>>>(bufA, cl1, N1, 32, x2);
    k_div_cnt<<<nb((long)N2 * 32), TB, 0, stream>>>(x2, cc1, N2, 32);
    hipMemsetAsync(pos2, 0, (size_t)N2 * 2 * 4, stream);
    k_scatter_feat<<<nb((long)N1 * 2), TB, 0, stream>>>(pos, cl1, N1, 2, pos2);
    k_div_cnt<<<nb((long)N2 * 2), TB, 0, stream>>>(pos2, cc1, N2, 2);
    k_map_edges<<<nb((long)2 * E), TB, 0, stream>>>(ei, cl1, (long)2 * E, ei2);
    hipMemsetAsync(red, 0, 4, stream);
    k_cart_d<<<nb(E), TB, 0, stream>>>(pos2, ei2, ei2 + E, E, ea2, red);
    k_cart_norm<<<nb((long)2 * E), TB, 0, stream>>>(ea2, red, (long)2 * E);

    // ======== conv2 (G-side, Cin=32 -> Cout=64; K = 800) ========
    hipMemsetAsync(bigG, 0, (size_t)N2 * 800 * 4, stream);
    hipMemsetAsync(cnt,  0, (size_t)N2 * 4, stream);
    k_edge_scatter_G<<<nb(E), TB, 0, stream>>>(x2, 32, 800, ei2, ei2 + E, ea2, E, bigG, cnt);
    k_pack_g<<<nb(25 * 32 * 64), TB, 0, stream>>>(W2, 32, 64, 800, wpk);
    k_gemm_wmma_f32<<<dim3(N2 / 64, 4), dim3(32, 4), 0, stream>>>(bigG, 800, wpk, agg, 64, 800);
    k_epilogue_elu<<<nb((long)N2 * 64), TB, 0, stream>>>(agg, cnt, x2, 32, r2, b2, bufA, N2, 64);

    // ======== pool level2 -> level3 ========
    hipMemsetAsync(cc2, 0, (size_t)N3 * 4, stream);
    k_count<<<nb(N2), TB, 0, stream>>>(cl2, N2, cc2);
    hipMemsetAsync(gs2, 0, (size_t)N3 * 32 * 4, stream);
    k_scatter_feat<<<nb((long)N2 * 32), TB, 0, stream>>>(x2, cl2, N2, 32, gs2);
    k_pweights<<<nb((long)N2 * 32), TB, 0, stream>>>(x2, cl2, N2, 32, cc2, gs2, w2b);
    hipMemsetAsync(x3, 0, (size_t)N3 * 64 * 4, stream);
    k_scatter_feat<<<nb((long)N2 * 64), TB, 0, stream>>>(bufA, cl2, N2, 64, x3);
    k_div_cnt<<<nb((long)N3 * 64), TB, 0, stream>>>(x3, cc2, N3, 64);
    hipMemsetAsync(pos3, 0, (size_t)N3 * 2 * 4, stream);
    k_scatter_feat<<<nb((long)N2 * 2), TB, 0, stream>>>(pos2, cl2, N2, 2, pos3);
    k_div_cnt<<<nb((long)N3 * 2), TB, 0, stream>>>(pos3, cc2, N3, 2);
    k_map_edges<<<nb((long)2 * E), TB, 0, stream>>>(ei2, cl2, (long)2 * E, ei3);
    hipMemsetAsync(red, 0, 4, stream);
    k_cart_d<<<nb(E), TB, 0, stream>>>(pos3, ei3, ei3 + E, E, ea3, red);
    k_cart_norm<<<nb((long)2 * E), TB, 0, stream>>>(ea3, red, (long)2 * E);

    // ======== conv3 (Z-side, Cin=64 -> Cout=32; Z[N3, 800]) ========
    hipMemsetAsync(wpk, 0, (size_t)800 * 64 * 4, stream);
    k_pack_z<<<nb(25 * 64 * 32), TB, 0, stream>>>(W3, 64, 32, wpk);
    k_gemm_wmma_f32<<<dim3(N3 / 64, 50), dim3(32, 4), 0, stream>>>(x3, 64, wpk, bigG, 800, 64);
    hipMemsetAsync(agg, 0, (size_t)N3 * 32 * 4, stream);
    hipMemsetAsync(cnt, 0, (size_t)N3 * 4, stream);
    k_edge_gather_Z<<<nb(E), TB, 0, stream>>>(bigG, 32, 800, ei3, ei3 + E, ea3, E, agg, cnt);
    k_epilogue_elu<<<nb((long)N3 * 32), TB, 0, stream>>>(agg, cnt, x3, 64, r3, b3, h3, N3, 32);

    // ======== upsample to level2, conv4 (Z-side, 32 -> 32; Z[N2, 800]) ========
    k_upsample<<<nb((long)N2 * 32), TB, 0, stream>>>(h3, cl2, w2b, 32, N2, 32, hb2);
    hipMemsetAsync(wpk, 0, (size_t)800 * 32 * 4, stream);
    k_pack_z<<<nb(25 * 32 * 32), TB, 0, stream>>>(W4, 32, 32, wpk);
    k_gemm_wmma_f32<<<dim3(N2 / 64, 50), dim3(32, 4), 0, stream>>>(hb2, 32, wpk, bigG, 800, 32);
    hipMemsetAsync(agg, 0, (size_t)N2 * 32 * 4, stream);
    hipMemsetAsync(cnt, 0, (size_t)N2 * 4, stream);
    k_edge_gather_Z<<<nb(E), TB, 0, stream>>>(bigG, 32, 800, ei2, ei2 + E, ea2, E, agg, cnt);
    k_epilogue_elu<<<nb((long)N2 * 32), TB, 0, stream>>>(agg, cnt, hb2, 32, r4, b4, h4, N2, 32);

    // ======== upsample to level1, conv5 (Z-side, 32 -> 1; Z cols padded 25 -> 32) ========
    k_upsample<<<nb((long)N1 * 32), TB, 0, stream>>>(h4, cl1, w1b, 1, N1, 32, bufA);
    hipMemsetAsync(red, 0, 4, stream);
    k_cart_d<<<nb(E), TB, 0, stream>>>(pos, ei, ei + E, E, ea1c, red);
    k_cart_norm<<<nb((long)2 * E), TB, 0, stream>>>(ea1c, red, (long)2 * E);
    hipMemsetAsync(wpk, 0, (size_t)32 * 32 * 4, stream);
    k_pack_z<<<nb(25 * 32 * 1), TB, 0, stream>>>(W5, 32, 1, wpk);
    k_gemm_wmma_f32<<<dim3(N1 / 64, 2), dim3(32, 4), 0, stream>>>(bufA, 32, wpk, bigG, 32, 32);
    hipMemsetAsync(agg, 0, (size_t)N1 * 4, stream);
    hipMemsetAsync(cnt, 0, (size_t)N1 * 4, stream);
    k_edge_gather_Z<<<nb(E), TB, 0, stream>>>(bigG, 1, 32, ei, ei + E, ea1c, E, agg, cnt);
    k_epilogue_elu<<<nb(N1), TB, 0, stream>>>(agg, cnt, bufA, 32, r5, b5, (float*)d_out, N1, 1);
}